// YOLOMaskLoss_68238440399481
// MI455X (gfx1250) — compile-verified
//
#include <hip/hip_runtime.h>
#include <math.h>

// Problem constants (shapes fixed by the reference's setup_inputs)
#define NB     4
#define AA     3
#define SSZ    13
#define NC     80
#define NMASK  32
#define GG     16
#define HH     104
#define WW     104
#define CELLS  (AA*SSZ*SSZ)     // 507
#define CP     512              // padded cells per batch
#define CCH    (5+NC+NMASK)     // 117 channels
#define HW     (HH*WW)
#define WTIL   7                // ceil(104/16) 16-wide pixel tiles per row
#define PTOT   (HH*WTIL)        // 728 pixel tiles per image

typedef float v2f __attribute__((ext_vector_type(2)));
typedef float v8f __attribute__((ext_vector_type(8)));

#if defined(__has_builtin)
#if __has_builtin(__builtin_amdgcn_wmma_f32_16x16x4_f32)
#define USE_WMMA_F32 1
#endif
#endif

__device__ __forceinline__ float bce_logits(float x, float z) {
    return fmaxf(x, 0.f) - x * z + log1pf(expf(-fabsf(x)));
}

// ---------------------------------------------------------------------------
// Kernel 1: one thread per (n, cell). Computes all scalar loss partials,
// tanh mask coefficients (for the WMMA GEMM), and seg-region metadata.
// ---------------------------------------------------------------------------
__global__ void yolo_cell_kernel(const float* __restrict__ preds,
                                 const float* __restrict__ target,
                                 const float* __restrict__ anchors,
                                 float* __restrict__ coeffs,
                                 int*   __restrict__ meta,
                                 float* __restrict__ pc)
{
    int tid  = blockIdx.x * blockDim.x + threadIdx.x;
    int n    = tid / CP;
    int cell = tid % CP;
    if (n >= NB) return;

    float* co = coeffs + (size_t)tid * NMASK;
    int*   mt = meta   + (size_t)tid * 8;
    float* p  = pc     + (size_t)tid * 8;

    if (cell >= CELLS) {  // padding cells: neutral everything (ws is poisoned)
        for (int k = 0; k < NMASK; k++) co[k] = 0.f;
        for (int q = 0; q < 8; q++) { mt[q] = 0; p[q] = 0.f; }
        mt[6] = 1;
        return;
    }

    int a  = cell / (SSZ * SSZ);
    int ij = cell - a * SSZ * SSZ;
    int i  = ij / SSZ;
    int j  = ij % SSZ;

    const float* pr = preds  + (size_t)(n * CELLS + cell) * CCH;
    const float* tg = target + (size_t)(n * CELLS + cell) * 7;

    float conf_t = tg[4];
    int obj   = (conf_t == 1.0f);
    int noobj = (conf_t == 0.0f);

    // mask coefficients for the seg GEMM
    for (int k = 0; k < NMASK; k++) co[k] = tanhf(pr[5 + NC + k]);

    // ---- segmentation region metadata ----
    float tx = tg[0], ty = tg[1], tw = tg[2], th = tg[3];
    float bx = (tx + (float)j) / (float)SSZ * (float)WW;
    float by = (ty + (float)i) / (float)SSZ * (float)HH;
    float bw = tw / (float)SSZ * (float)WW;
    float bh = th / (float)SSZ * (float)HH;
    int x1 = (int)floorf(bx - bw * 0.5f);
    int x2 = (int)floorf(bx + bw * 0.5f);
    int y1 = (int)floorf(by - bh * 0.5f);
    int y2 = (int)floorf(by + bh * 0.5f);
    int id = (int)tg[6];
    id = min(max(id, 0), GG - 1);
    int x1c = max(x1, 0), x2c = min(x2, WW);
    int y1c = max(y1, 0), y2c = min(y2, HH);
    int area = max(0, x2c - x1c) * max(0, y2c - y1c);
    int valid = (area > 0 && obj) ? 1 : 0;
    mt[0] = valid; mt[1] = x1c; mt[2] = x2c; mt[3] = y1c; mt[4] = y2c;
    mt[5] = id;    mt[6] = max(area, 1); mt[7] = 0;

    // ---- scalar loss partials ----
    float x = pr[4];  // conf logit
    p[0] = noobj ? bce_logits(x, 0.f) : 0.f;
    p[1] = noobj ? 1.f : 0.f;

    if (obj) {
        const float eps = 1e-9f;
        float sx = 1.f / (1.f + expf(-pr[0]));
        float sy = 1.f / (1.f + expf(-pr[1]));
        float aw = anchors[a * 2 + 0], ah = anchors[a * 2 + 1];
        float pbw = expf(pr[2]) * aw;
        float pbh = expf(pr[3]) * ah;
        // GIoU(pred_box=(sx,sy,pbw,pbh), tgt=(tx,ty,tw,th))
        float b1x1 = sx - pbw * 0.5f, b1y1 = sy - pbh * 0.5f;
        float b1x2 = sx + pbw * 0.5f, b1y2 = sy + pbh * 0.5f;
        float b2x1 = tx - tw * 0.5f,  b2y1 = ty - th * 0.5f;
        float b2x2 = tx + tw * 0.5f,  b2y2 = ty + th * 0.5f;
        float a1 = fmaxf(b1x2 - b1x1, 0.f) * fmaxf(b1y2 - b1y1, 0.f) + eps;
        float a2 = fmaxf(b2x2 - b2x1, 0.f) * fmaxf(b2y2 - b2y1, 0.f) + eps;
        float iw = fmaxf(fminf(b1x2, b2x2) - fmaxf(b1x1, b2x1), 0.f);
        float ih = fmaxf(fminf(b1y2, b2y2) - fmaxf(b1y1, b2y1), 0.f);
        float inter = iw * ih + eps;
        float uni = a1 + a2 - inter + eps;
        float iou = inter / uni;
        float cw = fmaxf(b1x2, b2x2) - fminf(b1x1, b2x1);
        float ch = fmaxf(b1y2, b2y2) - fminf(b1y1, b2y1);
        float carea = cw * ch + eps;
        float giou = iou - (carea - uni) / carea;

        float obj_tgt = conf_t * fmaxf(giou, 0.f);
        p[2] = bce_logits(x, obj_tgt);
        p[3] = 1.f;
        p[4] = bce_logits(sx, tx) + bce_logits(sy, ty);
        float twl = logf(1e-16f + tw / aw);
        float thl = logf(1e-16f + th / ah);
        float dw = pr[2] - twl, dh = pr[3] - thl;
        p[5] = dw * dw + dh * dh;
        p[6] = 1.f - giou;
        // class NLL via log-softmax over 80 classes
        int label = (int)tg[5];
        label = min(max(label, 0), NC - 1);
        float mx = pr[5];
        for (int c = 1; c < NC; c++) mx = fmaxf(mx, pr[5 + c]);
        float se = 0.f;
        for (int c = 0; c < NC; c++) se += expf(pr[5 + c] - mx);
        p[7] = (mx + logf(se)) - pr[5 + label];
    } else {
        p[2] = p[3] = p[4] = p[5] = p[6] = p[7] = 0.f;
    }
}

// ---------------------------------------------------------------------------
// Kernel 2: segmentation loss. One block per (n, 16-cell tile). Each wave
// owns pixel tiles (16 consecutive w in one row h), skips tiles outside the
// union bbox of valid cells, and computes inst = coeffs(16x32) * proto(32x16)
// with 8 chained V_WMMA_F32_16X16X4_F32, then masked BCE accumulation.
// B fragments come from direct lane-addressed global loads (no register-array
// indexing -> no v_cndmask cascades). Deterministic fixed-order reductions.
// ---------------------------------------------------------------------------
__global__ void yolo_seg_kernel(const float* __restrict__ proto,
                                const float* __restrict__ masks,
                                const float* __restrict__ coeffs,
                                const int*   __restrict__ meta,
                                float* __restrict__ segpart)
{
    int n     = blockIdx.x / 32;
    int mtile = blockIdx.x % 32;
    int t     = threadIdx.x;
    int lane  = t & 31;
    int wave  = t >> 5;
    int lanelo = lane & 15, lanehi = lane >> 4;

    __shared__ float sc[16][NMASK];
    __shared__ int sx1[16], sx2[16], sy1[16], sy2[16], sid[16], sval[16], sarea[16];
    __shared__ int ub[4];
    __shared__ float partial[8][16];
    __shared__ float cellsum[16];

    int cbase = n * CP + mtile * 16;
    if (t < 16) {
        const int* m = meta + (size_t)(cbase + t) * 8;
        sval[t] = m[0]; sx1[t] = m[1]; sx2[t] = m[2];
        sy1[t] = m[3]; sy2[t] = m[4]; sid[t] = m[5]; sarea[t] = m[6];
    }
    for (int idx = t; idx < 16 * NMASK; idx += blockDim.x) {
        sc[idx / NMASK][idx % NMASK] =
            coeffs[(size_t)(cbase + idx / NMASK) * NMASK + (idx % NMASK)];
    }
    __syncthreads();
    if (t == 0) {
        int uy1 = HH, uy2 = 0, ux1 = WW, ux2 = 0;
        for (int c = 0; c < 16; c++) if (sval[c]) {
            uy1 = min(uy1, sy1[c]); uy2 = max(uy2, sy2[c]);
            ux1 = min(ux1, sx1[c]); ux2 = max(ux2, sx2[c]);
        }
        ub[0] = uy1; ub[1] = uy2; ub[2] = ux1; ub[3] = ux2;
    }
    __syncthreads();
    int uy1 = ub[0], uy2 = ub[1], ux1 = ub[2], ux2 = ub[3];

    const float* Pn = proto + (size_t)n * NMASK * HW;
    const float* Mn = masks + (size_t)n * GG * HW;

#ifdef USE_WMMA_F32
    // A-matrix (16 cells x K=32), loop-invariant: ISA layout — lane holds row
    // M=lane%16; lanes 0-15 carry K pair (4kk+0,4kk+1), lanes 16-31 (4kk+2,4kk+3)
    v2f areg[8];
#pragma unroll
    for (int kk = 0; kk < 8; kk++) {
        int kb = kk * 4 + lanehi * 2;
        areg[kk].x = sc[lanelo][kb];
        areg[kk].y = sc[lanelo][kb + 1];
    }
#endif

    // Hoist per-(lane, r) cell metadata into registers (constant r indexing).
    // valid==0 is folded into impossible bounds so the inner test is compares only.
    int rx1[8], rx2[8], ry1[8], ry2[8];
    const float* rmask[8];
#pragma unroll
    for (int r = 0; r < 8; r++) {
        int m = r + lanehi * 8;
        int v = sval[m];
        rx1[r] = v ? sx1[m] : (1 << 20);
        rx2[r] = v ? sx2[m] : -(1 << 20);
        ry1[r] = v ? sy1[m] : (1 << 20);
        ry2[r] = v ? sy2[m] : -(1 << 20);
        rmask[r] = Mn + (size_t)sid[m] * HW;
    }

    float wsum[8];
#pragma unroll
    for (int r = 0; r < 8; r++) wsum[r] = 0.f;

    for (int pt = wave; pt < PTOT; pt += 8) {
        int h  = pt / WTIL;
        int w0 = (pt - h * WTIL) * 16;
        // wave-uniform early exit (EXEC stays all-ones for the WMMA)
        if (h < uy1 || h >= uy2 || w0 >= ux2 || w0 + 16 <= ux1) continue;

        int wcol = w0 + lanelo;
        // clamp address only; out-of-image columns produce finite garbage that
        // the region test (wcol < x2 <= 104) provably never consumes.
        int wc = min(wcol, WW - 1);

#ifdef USE_WMMA_F32
        // B fragment: lane supplies proto rows (4kk + lanehi*2, +1) at column wc.
        const float* bbase = Pn + (size_t)lanehi * 2 * HW + (size_t)h * WW + wc;
        v8f acc = {0.f, 0.f, 0.f, 0.f, 0.f, 0.f, 0.f, 0.f};
#pragma unroll
        for (int kk = 0; kk < 8; kk++) {
            v2f b;
            b.x = bbase[(size_t)(kk * 4) * HW];
            b.y = bbase[(size_t)(kk * 4 + 1) * HW];
            acc = __builtin_amdgcn_wmma_f32_16x16x4_f32(
                false, areg[kk], false, b, (short)0, acc, false, false);
        }
#else
        float pcol[NMASK];
#pragma unroll
        for (int k = 0; k < NMASK; k++)
            pcol[k] = Pn[(size_t)k * HW + h * WW + wc];
        float acc[8];
#pragma unroll
        for (int r = 0; r < 8; r++) {
            int m = r + lanehi * 8;
            float s = 0.f;
            for (int k = 0; k < NMASK; k++) s += sc[m][k] * pcol[k];
            acc[r] = s;
        }
#endif
        // D layout: VGPR r -> cell (r + lanehi*8), column = lanelo
#pragma unroll
        for (int r = 0; r < 8; r++) {
            if (h >= ry1[r] && h < ry2[r] && wcol >= rx1[r] && wcol < rx2[r]) {
                float z = rmask[r][(size_t)h * WW + wcol];
                wsum[r] += bce_logits(acc[r], z);
            }
        }
    }

    // reduce the 16 pixel-columns within each half-wave (fixed order)
#pragma unroll
    for (int r = 0; r < 8; r++) {
        for (int off = 1; off < 16; off <<= 1)
            wsum[r] += __shfl_xor(wsum[r], off, 16);
    }
    if (lanelo == 0) {
#pragma unroll
        for (int r = 0; r < 8; r++) partial[wave][r + lanehi * 8] = wsum[r];
    }
    __syncthreads();
    if (t < 16) {
        float tot = 0.f;
        for (int w = 0; w < 8; w++) tot += partial[w][t];
        cellsum[t] = sval[t] ? tot / (float)sarea[t] : 0.f;
    }
    __syncthreads();
    if (t == 0) {
        float s = 0.f;
        for (int c = 0; c < 16; c++) s += cellsum[c];
        segpart[blockIdx.x] = s;
    }
}

// ---------------------------------------------------------------------------
// Kernel 3: deterministic final reduction (single block) + loss assembly.
// ---------------------------------------------------------------------------
__global__ void yolo_reduce_kernel(const float* __restrict__ pc,
                                   const int*   __restrict__ meta,
                                   const float* __restrict__ segpart,
                                   float* __restrict__ out)
{
    __shared__ float red[256][10];
    int t = threadIdx.x;
    float loc[10];
    for (int q = 0; q < 10; q++) loc[q] = 0.f;
    for (int idx = t; idx < NB * CP; idx += 256) {
        const float* p = pc + (size_t)idx * 8;
        for (int q = 0; q < 8; q++) loc[q] += p[q];
        loc[8] += (float)meta[(size_t)idx * 8];  // valid count
    }
    for (int idx = t; idx < NB * 32; idx += 256) loc[9] += segpart[idx];
    for (int q = 0; q < 10; q++) red[t][q] = loc[q];
    __syncthreads();
    for (int s = 128; s > 0; s >>= 1) {
        if (t < s)
            for (int q = 0; q < 10; q++) red[t][q] += red[t + s][q];
        __syncthreads();
    }
    if (t == 0) {
        float noobj_sum = red[0][0], n_noobj = red[0][1];
        float obj_sum   = red[0][2], n_obj   = red[0][3];
        float xy_sum    = red[0][4], wh_sum  = red[0][5];
        float giou_sum  = red[0][6], nll_sum = red[0][7];
        float nvalid    = red[0][8], seg_sum = red[0][9];

        float noobj_loss = noobj_sum / n_noobj;
        float obj_loss   = obj_sum / n_obj;
        float box_loss   = xy_sum / (n_obj * 2.f) + wh_sum / (n_obj * 2.f)
                         + giou_sum / n_obj;
        float class_loss = nll_sum / n_obj;
        float seg_loss   = seg_sum / (nvalid + 1e-9f);

        float box_l = 8.f * box_loss;
        float obj_l = 2.f * obj_loss;
        float noobj_l = 4.f * noobj_loss;
        float cls_l = 1.f * class_loss;
        float seg_l = 10.f * seg_loss;
        out[0] = box_l; out[1] = obj_l; out[2] = noobj_l;
        out[3] = cls_l; out[4] = seg_l;
        out[5] = box_l + obj_l + noobj_l + cls_l + seg_l;
    }
}

extern "C" void kernel_launch(void* const* d_in, const int* in_sizes, int n_in,
                              void* d_out, int out_size, void* d_ws, size_t ws_size,
                              hipStream_t stream) {
    const float* preds   = (const float*)d_in[0];
    const float* target  = (const float*)d_in[1];
    const float* anchors = (const float*)d_in[2];
    const float* proto   = (const float*)d_in[3];
    const float* masks   = (const float*)d_in[4];
    float* out = (float*)d_out;

    char* ws = (char*)d_ws;
    size_t off0 = 0;                                        // coeffs
    size_t off1 = off0 + sizeof(float) * NB * CP * NMASK;   // meta
    size_t off2 = off1 + sizeof(int)   * NB * CP * 8;       // per-cell partials
    size_t off3 = off2 + sizeof(float) * NB * CP * 8;       // seg partials
    float* coeffs  = (float*)(ws + off0);
    int*   meta    = (int*)  (ws + off1);
    float* pc      = (float*)(ws + off2);
    float* segpart = (float*)(ws + off3);

    yolo_cell_kernel<<<(NB * CP) / 256, 256, 0, stream>>>(
        preds, target, anchors, coeffs, meta, pc);
    yolo_seg_kernel<<<NB * 32, 256, 0, stream>>>(
        proto, masks, coeffs, meta, segpart);
    yolo_reduce_kernel<<<1, 256, 0, stream>>>(pc, meta, segpart, out);
}